// dynamics_rootfind_53403623358903
// MI455X (gfx1250) — compile-verified
//
#include <hip/hip_runtime.h>
#include <math.h>

// ---------------------------------------------------------------------------
// Block = 256 threads = 8 wave32. Block tile 64(M) x 64(N), BK = 16.
// Wave (wm 0..3, wn 0..1) owns a 16x32 output tile = 2 f32 WMMA accumulators.
// Global -> LDS staging via GLOBAL_LOAD_ASYNC_TO_LDS (ASYNCcnt), double-buffered,
// steady-state loop peeled so the hot body is branch-free.
// ---------------------------------------------------------------------------
#define BM 64
#define BN 64
#define BK 16
#define PITCH 20  // floats/row: 80B rows -> 16B-aligned b128 stores, conflict-free frags

typedef float v2f __attribute__((ext_vector_type(2)));
typedef float v8f __attribute__((ext_vector_type(8)));

__device__ __forceinline__ v8f wmma_f32x4(v2f a, v2f b, v8f c) {
  // 8 args: (neg_a, A, neg_b, B, c_mod, C, reuse_a, reuse_b)
  return __builtin_amdgcn_wmma_f32_16x16x4_f32(
      false, a, false, b, (short)0, c, false, false);
}

// Flat pointer to a __shared__ object: low 32 bits are the LDS byte offset
// (ISA 10.2: LDS aperture maps addr[31:0] -> LDS_ADDR).
__device__ __forceinline__ unsigned lds_off(const void* p) {
  return (unsigned)(size_t)p;
}

__device__ __forceinline__ void async_ld_b128(unsigned lds, const float* g) {
  asm volatile("global_load_async_to_lds_b128 %0, %1, off"
               :: "v"(lds), "v"(g) : "memory");
}
__device__ __forceinline__ void async_ld_b32(unsigned lds, const float* g) {
  asm volatile("global_load_async_to_lds_b32 %0, %1, off"
               :: "v"(lds), "v"(g) : "memory");
}

// EPI == 0 : C = tanh(acc + bias[col])  (store to Cout)
// EPI == 1 : rowacc[row] += sum_n acc[row][n]^2  (atomic, no C store)
// TRANSB==1: logical B[k][n] = Bmat[n*K + k]   (x @ W2^T)
// TRANSB==0: logical B[k][n] = Bmat[k*N + n]   (@ Wv, row-major)
template <int TRANSB, int EPI>
__global__ __launch_bounds__(256)
void gemm_wmma_f32_kernel(const float* __restrict__ A,
                          const float* __restrict__ Bmat,
                          const float* __restrict__ bias,
                          float* __restrict__ Cout,
                          float* __restrict__ rowacc,
                          int M, int N, int K) {
  // Double-buffered tiles, both fragment-friendly:
  //   As[m][k] (m-major), Bs[n][k] (n-major) -> frags are 2 consecutive floats.
  __shared__ float As[2][BM * PITCH];
  __shared__ float Bs[2][BN * PITCH];

  const int tid   = threadIdx.x;
  const int lane  = tid & 31;
  const int wave  = tid >> 5;     // 0..7
  const int wm    = wave & 3;     // 16-row sub-tile
  const int wn    = wave >> 2;    // 32-col half
  const int lhalf = lane >> 4;    // 0 | 1
  const int lmod  = lane & 15;

  const int bm0 = blockIdx.x * BM;
  const int bn0 = blockIdx.y * BN;

  v8f c0 = {0.f, 0.f, 0.f, 0.f, 0.f, 0.f, 0.f, 0.f};
  v8f c1 = c0;

  // ---- async staging of one BK-chunk into buffer `buf` ----
  auto stage = [&](int buf, int k0) {
    {  // A tile: 64 rows x 16 k, one b128 per thread (1 async instr / wave)
      const int m  = tid >> 2;
      const int kq = (tid & 3) * 4;
      async_ld_b128(lds_off(&As[buf][m * PITCH + kq]),
                    &A[(size_t)(bm0 + m) * K + (k0 + kq)]);
    }
    if (TRANSB) {  // W2 rows are k-contiguous: one b128 per thread
      const int n  = tid >> 2;
      const int kq = (tid & 3) * 4;
      async_ld_b128(lds_off(&Bs[buf][n * PITCH + kq]),
                    &Bmat[(size_t)(bn0 + n) * K + (k0 + kq)]);
    } else {       // Wv row-major: n-contiguous in memory, k-scatter in LDS
      const int n  = tid & 63;
      const int kb = tid >> 6;  // 0..3
#pragma unroll
      for (int i = 0; i < 4; ++i) {
        const int k = kb + i * 4;
        async_ld_b32(lds_off(&Bs[buf][n * PITCH + k]),
                     &Bmat[(size_t)(k0 + k) * N + (bn0 + n)]);
      }
    }
  };
  // async instrs per wave per stage: TRANSB: 2   NN: 5

  auto compute = [&](int buf) {
    const float* as = As[buf];
    const float* bs = Bs[buf];
#pragma unroll
    for (int kk = 0; kk < BK; kk += 4) {
      const int ka = kk + lhalf * 2;  // even -> 8B-aligned ds_load_b64
      v2f a  = *(const v2f*)&as[(wm * 16 + lmod) * PITCH + ka];
      v2f b0 = *(const v2f*)&bs[(wn * 32 + lmod) * PITCH + ka];
      v2f b1 = *(const v2f*)&bs[(wn * 32 + 16 + lmod) * PITCH + ka];
      c0 = wmma_f32x4(a, b0, c0);
      c1 = wmma_f32x4(a, b1, c1);
    }
  };

  stage(0, 0);
  int cur = 0;
  // Steady state: always a next chunk to prefetch -> branch-free body.
  for (int k0 = 0; k0 < K - BK; k0 += BK) {
    stage(cur ^ 1, k0 + BK);
    // Async loads complete in order: waiting to <= (just-issued count)
    // guarantees the current buffer's data has landed in LDS.
    if (TRANSB) asm volatile("s_wait_asynccnt 0x2" ::: "memory");
    else        asm volatile("s_wait_asynccnt 0x5" ::: "memory");
    __syncthreads();          // all waves' staging visible
    compute(cur);
    __syncthreads();          // done reading `cur` before it is overwritten
    cur ^= 1;
  }
  // Peeled last chunk: nothing left to prefetch.
  asm volatile("s_wait_asynccnt 0x0" ::: "memory");
  __syncthreads();
  compute(cur);

  // ---- epilogue (C/D layout: VGPR v -> row v + 8*lhalf, lane -> col lmod) ----
  const int row0    = bm0 + wm * 16;
  const int colbase = bn0 + wn * 32;

  if (EPI == 0) {
    const float bv0 = bias[colbase + lmod];
    const float bv1 = bias[colbase + 16 + lmod];
#pragma unroll
    for (int v = 0; v < 8; ++v) {
      const int r = row0 + v + lhalf * 8;
      Cout[(size_t)r * N + colbase + lmod]      = tanhf(c0[v] + bv0);
      Cout[(size_t)r * N + colbase + 16 + lmod] = tanhf(c1[v] + bv1);
    }
  } else {
#pragma unroll
    for (int v = 0; v < 8; ++v) {
      float s = c0[v] * c0[v] + c1[v] * c1[v];
      s += __shfl_xor(s, 1, 16);
      s += __shfl_xor(s, 2, 16);
      s += __shfl_xor(s, 4, 16);
      s += __shfl_xor(s, 8, 16);
      if (lmod == 0)
        atomicAdd(&rowacc[row0 + v + lhalf * 8], s);
    }
  }
}

__global__ void zero_f32_kernel(float* __restrict__ p, int n) {
  const int i = blockIdx.x * blockDim.x + threadIdx.x;
  if (i < n) p[i] = 0.f;
}

// Per row: Newton on alpha (V(a*f) = a^2*Vf exactly), then out = alpha * F.
__global__ __launch_bounds__(256)
void alpha_scale_kernel(const float* __restrict__ F,
                        const float* __restrict__ Vx,
                        const float* __restrict__ Vf,
                        float* __restrict__ out, int N) {
  __shared__ float s_alpha;
  const int b = blockIdx.x;
  if (threadIdx.x == 0) {
    const float vf  = Vf[b];
    const float tgt = 0.99f * Vx[b];
    float a = 1.0f;
    for (int it = 0; it < 64; ++it) {
      const float g = a * a * vf - tgt;
      if (!(g > 1e-3f)) break;  // matches while_loop condition (incl. a=1 early exit)
      a = a - g / (2.0f * a * vf);
    }
    s_alpha = a;
  }
  __syncthreads();
  const float a = s_alpha;
  const float4* f4 = (const float4*)(F + (size_t)b * N);
  float4*       o4 = (float4*)(out + (size_t)b * N);
  for (int i = threadIdx.x; i < N / 4; i += blockDim.x) {
    float4 v = f4[i];
    v.x *= a; v.y *= a; v.z *= a; v.w *= a;
    o4[i] = v;
  }
}

extern "C" void kernel_launch(void* const* d_in, const int* in_sizes, int n_in,
                              void* d_out, int out_size, void* d_ws, size_t ws_size,
                              hipStream_t stream) {
  // Inputs: 0:x [B,N]  1:W1 (dead)  2:b1 (dead)  3:W2 [N,N]  4:b2 [N]  5:Wv [N,N]
  const float* x  = (const float*)d_in[0];
  const float* W2 = (const float*)d_in[3];
  const float* b2 = (const float*)d_in[4];
  const float* Wv = (const float*)d_in[5];
  float* out = (float*)d_out;

  const int N = in_sizes[4];          // 1024
  const int B = in_sizes[0] / N;      // 8192

  // Workspace: F [B*N] f32, then Vx [B], Vf [B]
  float* F  = (float*)d_ws;
  float* Vx = F + (size_t)B * N;
  float* Vf = Vx + B;

  dim3 blk(256);
  dim3 grid(B / BM, N / BN);

  // Vx/Vf accumulated atomically -> zero them every call (ws is not re-zeroed)
  zero_f32_kernel<<<(2 * B + 255) / 256, blk, 0, stream>>>(Vx, 2 * B);

  // F = tanh(x @ W2^T + b2)
  gemm_wmma_f32_kernel<1, 0><<<grid, blk, 0, stream>>>(x, W2, b2, F, nullptr, B, N, N);
  // Vx[b] = || x[b] @ Wv ||^2
  gemm_wmma_f32_kernel<0, 1><<<grid, blk, 0, stream>>>(x, Wv, nullptr, nullptr, Vx, B, N, N);
  // Vf[b] = || F[b] @ Wv ||^2
  gemm_wmma_f32_kernel<0, 1><<<grid, blk, 0, stream>>>(F, Wv, nullptr, nullptr, Vf, B, N, N);
  // out = alpha[b] * F[b,:]
  alpha_scale_kernel<<<B, blk, 0, stream>>>(F, Vx, Vf, out, N);
}